// GRAPHSAGE_5403068859076
// MI455X (gfx1250) — compile-verified
//
#include <hip/hip_runtime.h>
#include <hip/hip_bf16.h>

typedef float v2f __attribute__((ext_vector_type(2)));
typedef float v8f __attribute__((ext_vector_type(8)));

#define NNODES 50000
#define NEDGES 800000
#define DIN_   64
#define DH_    128
#define MH_    200
#define EPS_   1e-5f

// ---------------------------------------------------------------- zero
__global__ __launch_bounds__(256) void zero_kernel(float4* __restrict__ p, int n4) {
    int i = blockIdx.x * blockDim.x + threadIdx.x;
    int stride = gridDim.x * blockDim.x;
    float4 z = make_float4(0.f, 0.f, 0.f, 0.f);
    for (; i < n4; i += stride) p[i] = z;
}

// ---------------------------------------------------------------- degree
__global__ __launch_bounds__(256) void degree_kernel(const int* __restrict__ dst,
                                                     float* __restrict__ deg) {
    int e = blockIdx.x * blockDim.x + threadIdx.x;
    if (e < NEDGES) atomicAdd(&deg[dst[e]], 1.0f);
}

// ---------------------------------------------------------------- scatter-add msg[dst] += H[src]
template <int D>
__global__ __launch_bounds__(256) void scatter_kernel(const float* __restrict__ H,
                                                      const int* __restrict__ src,
                                                      const int* __restrict__ dst,
                                                      float* __restrict__ msg) {
    constexpr int CH = D / 4;                 // float4 chunks per edge
    int tid = blockIdx.x * blockDim.x + threadIdx.x;
    if (tid >= NEDGES * CH) return;
    int e = tid / CH;
    int c = tid % CH;
    int s = src[e];
    int d = dst[e];
    const float4 v = *reinterpret_cast<const float4*>(&H[(size_t)s * D + c * 4]);
    float* base = &msg[(size_t)d * D + c * 4];
    atomicAdd(base + 0, v.x);
    atomicAdd(base + 1, v.y);
    atomicAdd(base + 2, v.z);
    atomicAdd(base + 3, v.w);
}

// ---------------------------------------------------------------- fused (normalize) + GEMM + bias + relu
// OUT[N, NCOLS] = relu( A[N, K] @ W[K, NCOLS] + bias )
// NORM: A = (Xmsg + 2*Xself) / (deg + 2);  else A = Xmsg
template <int K, int NCOLS, bool NORM>
__global__ __launch_bounds__(256) void gemm_tile_kernel(const float* __restrict__ Xmsg,
                                                        const float* __restrict__ Xself,
                                                        const float* __restrict__ deg,
                                                        const float* __restrict__ W,
                                                        const float* __restrict__ bias,
                                                        float* __restrict__ OUT) {
    constexpr int NT = (NCOLS + 15) / 16;     // column tiles of 16
    constexpr int STRIDE = K + 4;             // even -> 8B-aligned float2 LDS reads
    __shared__ __align__(16) float As[16 * STRIDE];

    const int row0 = blockIdx.x * 16;         // 50000 = 3125 * 16, no row guard needed
    const int tid  = threadIdx.x;
    const int wave = tid >> 5;
    const int lane = tid & 31;

    // stage normalized A tile (16 x K) into LDS, coalesced
    for (int i = tid; i < 16 * K; i += 256) {
        const int r = i / K;
        const int c = i % K;
        const int g = row0 + r;
        float val;
        if (NORM) {
            const float m = Xmsg[(size_t)g * K + c];
            const float s = Xself[(size_t)g * K + c];
            val = (m + 2.0f * s) / (deg[g] + 2.0f);
        } else {
            val = Xmsg[(size_t)g * K + c];
        }
        As[r * STRIDE + c] = val;
    }
    __syncthreads();

    const int m  = lane & 15;                 // A row this lane supplies
    const int kh = lane >> 4;                 // 0: K={0,1}, 1: K={2,3} inside the 4-step
    const int n  = lane & 15;                 // B/C column this lane supplies

    for (int t = wave; t < NT; t += 8) {
        const int col0 = t * 16;
        const int cn = col0 + n;
        const bool colok = (NCOLS % 16 == 0) || (cn < NCOLS);

        v8f acc = {};
        #pragma unroll
        for (int ks = 0; ks < K / 4; ++ks) {
            const int kk = 4 * ks + 2 * kh;
            // A: 16x4 f32 -> lane holds A[m][kk], A[m][kk+1]
            v2f a = *reinterpret_cast<const v2f*>(&As[m * STRIDE + kk]);
            // B: 4x16 f32 -> lane holds W[kk][cn], W[kk+1][cn]
            v2f b;
            b.x = colok ? W[(size_t)kk * NCOLS + cn] : 0.0f;
            b.y = colok ? W[(size_t)(kk + 1) * NCOLS + cn] : 0.0f;
            acc = __builtin_amdgcn_wmma_f32_16x16x4_f32(
                /*neg_a=*/false, a, /*neg_b=*/false, b,
                /*c_mod=*/(short)0, acc, /*reuse_a=*/false, /*reuse_b=*/false);
        }

        const float bv = colok ? bias[cn] : 0.0f;
        #pragma unroll
        for (int v = 0; v < 8; ++v) {
            // C/D layout: VGPR v, lanes 0-15 -> M=v, lanes 16-31 -> M=v+8
            const int mrow = v + 8 * kh;
            float x = acc[v] + bv;
            x = fmaxf(x, 0.0f);               // all three GEMMs are followed by ReLU
            if (colok) OUT[(size_t)(row0 + mrow) * NCOLS + cn] = x;
        }
    }
}

// ---------------------------------------------------------------- BN column stats (sum, sumsq)
__global__ __launch_bounds__(256) void bn_stats_kernel(const float* __restrict__ z,
                                                       float* __restrict__ stats) {
    const int col = threadIdx.x;
    if (col >= MH_) return;
    const int rows_per_block = NNODES / 200;  // 250, gridDim.x == 200
    const int r0 = blockIdx.x * rows_per_block;
    float s = 0.f, s2 = 0.f;
    for (int r = 0; r < rows_per_block; ++r) {
        const float v = z[(size_t)(r0 + r) * MH_ + col];
        s += v;
        s2 += v * v;
    }
    atomicAdd(&stats[col], s);
    atomicAdd(&stats[MH_ + col], s2);
}

// ---------------------------------------------------------------- BN apply + final dot + sigmoid
__global__ __launch_bounds__(256) void final_kernel(const float* __restrict__ z,
                                                    const float* __restrict__ stats,
                                                    const float* __restrict__ gamma,
                                                    const float* __restrict__ beta,
                                                    const float* __restrict__ Wm2,
                                                    const float* __restrict__ bm2,
                                                    float* __restrict__ out) {
    const int lane = threadIdx.x & 31;
    const int node = blockIdx.x * 8 + (threadIdx.x >> 5);   // one wave32 per node
    if (node >= NNODES) return;
    const float invN = 1.0f / (float)NNODES;
    float partial = 0.f;
    for (int c = lane; c < MH_; c += 32) {
        const float mu  = stats[c] * invN;
        const float var = stats[MH_ + c] * invN - mu * mu;
        const float zn  = (z[(size_t)node * MH_ + c] - mu) * rsqrtf(var + EPS_) * gamma[c] + beta[c];
        partial += zn * Wm2[c];
    }
    #pragma unroll
    for (int off = 16; off > 0; off >>= 1) partial += __shfl_xor(partial, off, 32);
    if (lane == 0) out[node] = 1.0f / (1.0f + expf(-(partial + bm2[0])));
}

// ---------------------------------------------------------------- launch
extern "C" void kernel_launch(void* const* d_in, const int* in_sizes, int n_in,
                              void* d_out, int out_size, void* d_ws, size_t ws_size,
                              hipStream_t stream) {
    const float* features = (const float*)d_in[0];
    const float* W1   = (const float*)d_in[1];
    const float* b1   = (const float*)d_in[2];
    const float* W2   = (const float*)d_in[3];
    const float* b2   = (const float*)d_in[4];
    const float* Wm1  = (const float*)d_in[5];
    const float* bm1  = (const float*)d_in[6];
    const float* gamma= (const float*)d_in[7];
    const float* beta = (const float*)d_in[8];
    const float* Wm2  = (const float*)d_in[9];
    const float* bm2  = (const float*)d_in[10];
    const int*   src  = (const int*)d_in[11];
    const int*   dst  = (const int*)d_in[12];
    float* out = (float*)d_out;

    float* ws    = (float*)d_ws;
    float* deg   = ws;                               // N
    float* msg   = deg + NNODES;                     // N*128 (layer1 uses first N*64)
    float* h1    = msg + (size_t)NNODES * DH_;       // N*128
    float* h2    = h1  + (size_t)NNODES * DH_;       // N*128
    float* z     = h2  + (size_t)NNODES * DH_;       // N*200
    float* stats = z   + (size_t)NNODES * MH_;       // 512 (sum[200], sumsq[200], pad)

    const int T = 256;

    // 1. zero deg+msg (contiguous) and stats
    {
        int n4 = (NNODES + NNODES * DH_) / 4;
        zero_kernel<<<2048, T, 0, stream>>>((float4*)deg, n4);
        zero_kernel<<<1, 128, 0, stream>>>((float4*)stats, 128);
    }
    // 2. in-degree
    degree_kernel<<<(NEDGES + T - 1) / T, T, 0, stream>>>(dst, deg);
    // 3. layer-1 message scatter (E * 16 threads)
    scatter_kernel<DIN_><<<(NEDGES * (DIN_ / 4) + T - 1) / T, T, 0, stream>>>(features, src, dst, msg);
    // 4. h1 = relu(((msg + 2f)/(deg+2)) @ W1 + b1)
    gemm_tile_kernel<DIN_, DH_, true><<<NNODES / 16, T, 0, stream>>>(msg, features, deg, W1, b1, h1);
    // 5. re-zero msg for layer 2
    zero_kernel<<<2048, T, 0, stream>>>((float4*)msg, NNODES * DH_ / 4);
    // 6. layer-2 message scatter (E * 32 threads)
    scatter_kernel<DH_><<<(NEDGES * (DH_ / 4) + T - 1) / T, T, 0, stream>>>(h1, src, dst, msg);
    // 7. h2 = relu(((msg + 2h1)/(deg+2)) @ W2 + b2)
    gemm_tile_kernel<DH_, DH_, true><<<NNODES / 16, T, 0, stream>>>(msg, h1, deg, W2, b2, h2);
    // 8. z = relu(h2 @ Wm1 + bm1)
    gemm_tile_kernel<DH_, MH_, false><<<NNODES / 16, T, 0, stream>>>(h2, h2, deg, Wm1, bm1, z);
    // 9. BN batch statistics
    bn_stats_kernel<<<200, T, 0, stream>>>(z, stats);
    // 10. BN apply + Wm2 dot + sigmoid
    final_kernel<<<(NNODES + 7) / 8, T, 0, stream>>>(z, stats, gamma, beta, Wm2, bm2, out);
}